// Coboundary_1580547969961
// MI455X (gfx1250) — compile-verified
//
#include <hip/hip_runtime.h>

// ---------------------------------------------------------------------------
// y[b,o,n] = bias[o] + sum_c theta[o,c] * sum_m D[n,m] * x[b,c,m]
// Folded:  w[bo,m] = sum_c theta[o,c]*x[b,c,m]   (bo = b*8+o, 16 rows)
//          y[bo,n] = bias[o] + sum_m D[n,m] * w[bo,m]
// -> single GEMM  C(16384 x 16) = D(16384 x 16384) * w^T, tiled with
//    V_WMMA_F32_16X16X4_F32 (C tile = 16 n-rows x 16 bo-cols).
//
// Roofline: D = 1.07 GB streamed exactly once @ 23.3 TB/s => ~46 us floor.
// 8.6 GFLOP fp32 over that window (~187 TFLOP/s) sits well under the f32
// WMMA ceiling, so the kernel is built around HBM streaming:
//  - one nontemporal b128 per lane feeds TWO wmmas (k->slot permutation,
//    applied identically to A and B, so slot products still pair correctly)
//  - each wave carries TWO n-tiles sharing one B stream (w L2 traffic /2)
//  - D loads are TH=NT so the streamed 1 GiB never evicts the 1 MB hot w.
// ---------------------------------------------------------------------------

typedef float v2f __attribute__((ext_vector_type(2)));
typedef float v4f __attribute__((ext_vector_type(4)));
typedef float v8f __attribute__((ext_vector_type(8)));

constexpr int N    = 16384;
constexpr int M    = 16384;
constexpr int BB   = 2;
constexpr int CIN  = 4;
constexpr int COUT = 8;
constexpr int BO   = BB * COUT;   // 16 fused output rows == WMMA N dim
constexpr int NT   = 2;           // n-tiles per wave (B-stream reuse)
constexpr int WAVES_PER_BLOCK = 8;
constexpr int KSPLIT = 8;         // K-parallel split, accumulated via f32 atomics
constexpr int KC = M / KSPLIT;    // 2048 k per block

// ---- kernel 1: fold theta into x -> w[bo][m] (fp32, row pitch M) ----------
__global__ void fold_theta_kernel(const float* __restrict__ x,
                                  const float* __restrict__ theta,
                                  float* __restrict__ w) {
    int m = blockIdx.x * blockDim.x + threadIdx.x;
    if (m >= M) return;
    float xv[BB][CIN];
#pragma unroll
    for (int b = 0; b < BB; ++b)
#pragma unroll
        for (int c = 0; c < CIN; ++c)
            xv[b][c] = x[(size_t)(b * CIN + c) * M + m];
#pragma unroll
    for (int b = 0; b < BB; ++b)
#pragma unroll
        for (int o = 0; o < COUT; ++o) {
            float acc = 0.f;
#pragma unroll
            for (int c = 0; c < CIN; ++c)
                acc += theta[o * CIN + c] * xv[b][c];
            w[(size_t)(b * COUT + o) * M + m] = acc;
        }
}

// ---- kernel 2: init y with bias (harness poisons d_out) -------------------
__global__ void init_y_kernel(const float* __restrict__ bias,
                              float* __restrict__ y) {
    int i = blockIdx.x * blockDim.x + threadIdx.x;
    if (i < BO * N) {
        int bo = i / N;               // bo = b*COUT + o
        y[i] = bias[bo % COUT];
    }
}

// ---- kernel 3: main WMMA GEMM ---------------------------------------------
// grid = (N/(16*NT*WAVES_PER_BLOCK), KSPLIT), block = 256 (8 waves).
// k->slot permutation per wmma pair over an 8-wide k octet:
//   wmma0 slots: half0 -> m=k,k+1   ; half1 -> m=k+4,k+5
//   wmma1 slots: half0 -> m=k+2,k+3 ; half1 -> m=k+6,k+7
// so ONE b128 at (base + k + 4*half) feeds both wmmas of the octet, for A
// (D rows) and B (w rows) alike. Lane r and lane r+16 jointly consume row r
// contiguously -> full cacheline utilization while streaming.
__global__ __launch_bounds__(WAVES_PER_BLOCK * 32)
void coboundary_wmma_kernel(const float* __restrict__ D,
                            const float* __restrict__ w,
                            float* __restrict__ y) {
    const int lane = threadIdx.x & 31;
    const int wave = threadIdx.x >> 5;
    const int row  = lane & 15;
    const int half = lane >> 4;

    const int n0 = (blockIdx.x * WAVES_PER_BLOCK + wave) * (16 * NT);
    const int k0 = blockIdx.y * KC;

    // 16B-aligned: k0 multiple of 2048, 4*half in {0,4}
    const float* aPtr0 = D + (size_t)(n0 + row) * M + k0 + 4 * half;        // tile 0
    const float* aPtr1 = D + (size_t)(n0 + 16 + row) * M + k0 + 4 * half;   // tile 1
    const float* bPtr  = w + (size_t)row * M + k0 + 4 * half;               // shared B

    v8f c0 = {0.f, 0.f, 0.f, 0.f, 0.f, 0.f, 0.f, 0.f};
    v8f c1 = {0.f, 0.f, 0.f, 0.f, 0.f, 0.f, 0.f, 0.f};

#pragma unroll 4
    for (int k = 0; k < KC; k += 8) {
        // D: streamed exactly once -> nontemporal (keep L2 for w)
        v4f a0 = __builtin_nontemporal_load((const v4f*)(aPtr0 + k));
        v4f a1 = __builtin_nontemporal_load((const v4f*)(aPtr1 + k));
        v4f b  = *(const v4f*)(bPtr + k);   // hot 1 MB, L2-resident

        v2f aA0 = __builtin_shufflevector(a0, a0, 0, 1);
        v2f aB0 = __builtin_shufflevector(a0, a0, 2, 3);
        v2f aA1 = __builtin_shufflevector(a1, a1, 0, 1);
        v2f aB1 = __builtin_shufflevector(a1, a1, 2, 3);
        v2f bA  = __builtin_shufflevector(b,  b,  0, 1);
        v2f bB  = __builtin_shufflevector(b,  b,  2, 3);

        c0 = __builtin_amdgcn_wmma_f32_16x16x4_f32(false, aA0, false, bA,
                                                   (short)0, c0, false, false);
        c1 = __builtin_amdgcn_wmma_f32_16x16x4_f32(false, aA1, false, bA,
                                                   (short)0, c1, false, false);
        c0 = __builtin_amdgcn_wmma_f32_16x16x4_f32(false, aB0, false, bB,
                                                   (short)0, c0, false, false);
        c1 = __builtin_amdgcn_wmma_f32_16x16x4_f32(false, aB1, false, bB,
                                                   (short)0, c1, false, false);
    }

    // C layout: vgpr r, lanes0-15 -> (n = base+r,   bo = lane)
    //                   lanes16-31 -> (n = base+r+8, bo = lane-16)
#pragma unroll
    for (int r = 0; r < 8; ++r) {
        int n = n0 + r + 8 * half;
        atomicAdd(&y[(size_t)row * N + n],      c0[r]);  // global_atomic_add_f32
        atomicAdd(&y[(size_t)row * N + n + 16], c1[r]);
    }
}

// ---------------------------------------------------------------------------
extern "C" void kernel_launch(void* const* d_in, const int* in_sizes, int n_in,
                              void* d_out, int out_size, void* d_ws, size_t ws_size,
                              hipStream_t stream) {
    const float* D     = (const float*)d_in[0];   // (N, M)
    const float* x     = (const float*)d_in[1];   // (B, C_IN, M)
    const float* theta = (const float*)d_in[2];   // (C_OUT, C_IN)
    const float* bias  = (const float*)d_in[3];   // (1, C_OUT, 1)
    float* y = (float*)d_out;                     // (B, C_OUT, N) flat
    float* w = (float*)d_ws;                      // BO * M floats = 1 MiB

    fold_theta_kernel<<<(M + 255) / 256, 256, 0, stream>>>(x, theta, w);
    init_y_kernel<<<(BO * N + 255) / 256, 256, 0, stream>>>(bias, y);

    dim3 grid(N / (16 * NT * WAVES_PER_BLOCK), KSPLIT);
    coboundary_wmma_kernel<<<grid, WAVES_PER_BLOCK * 32, 0, stream>>>(D, w, y);
}